// VisualKGModule_11647951307142
// MI455X (gfx1250) — compile-verified
//
#include <hip/hip_runtime.h>

namespace {

constexpr int kNV   = 8192;
constexpr int kNCAM = 20;
constexpr int kCIN  = 2048;
constexpr int kH    = 4;
constexpr int kCOUT = 2048;
constexpr int kE    = 16384;
constexpr int kNHC  = kH * kCOUT;   // 8192 (W columns)
constexpr int kNJ   = kH * kNCAM;   // 80  (contraction dim of final GEMM)
constexpr float kNegSlope = 0.2f;

// workspace layout (in floats)
constexpr size_t OFF_HS    = 0;                                // Hs[80][2048]
constexpr size_t OFF_WATT  = OFF_HS    + (size_t)kNJ * kCOUT;  // W_att[2048][4]
constexpr size_t OFF_AD    = OFF_WATT  + (size_t)kCIN * kH;    // a_d[8192][4]
constexpr size_t OFF_AS    = OFF_AD    + (size_t)kNV * kH;     // a_s[20][4] (padded)
constexpr size_t OFF_MENC  = OFF_AS    + 128;                  // mEnc[8192][4] (uint)
constexpr size_t OFF_DEN   = OFF_MENC  + (size_t)kNV * kH;     // denom[8192][4]
constexpr size_t OFF_S     = OFF_DEN   + (size_t)kNV * kH;     // S[8192][80]
constexpr size_t OFF_CTPAD = OFF_S     + (size_t)kNV * kNJ;    // CT_pad[32][2048]

typedef float v2f __attribute__((ext_vector_type(2)));
typedef float v8f __attribute__((ext_vector_type(8)));

__device__ __forceinline__ float wave_sum(float v) {
#pragma unroll
  for (int off = 16; off > 0; off >>= 1) v += __shfl_down(v, off, 32);
  return v;
}

// order-preserving float<->uint map so unsigned atomicMax == float max.
// memset-0 pattern decodes below every finite float.
__device__ __forceinline__ unsigned f2ord(float f) {
  unsigned u = __float_as_uint(f);
  return (u & 0x80000000u) ? ~u : (u | 0x80000000u);
}
__device__ __forceinline__ float ord2f(unsigned e) {
  return __uint_as_float((e & 0x80000000u) ? (e ^ 0x80000000u) : ~e);
}

// ---- 1. Fold att_dst into W:  W_att[k][h] = sum_c W[k][h*2048+c]*att_dst[h][c]
// One wave per (k,h); float4 loads -> 16-byte vmem ops on the 67MB W stream.
__global__ void k_fold_att(const float* __restrict__ W,
                           const float* __restrict__ att_dst,
                           float* __restrict__ W_att) {
  int wave = (blockIdx.x * blockDim.x + threadIdx.x) >> 5;  // (k,h) pair
  int lane = threadIdx.x & 31;
  if (wave >= kCIN * kH) return;
  int k = wave >> 2, h = wave & 3;
  const float4* wrow = reinterpret_cast<const float4*>(W + (size_t)k * kNHC + (size_t)h * kCOUT);
  const float4* arow = reinterpret_cast<const float4*>(att_dst + (size_t)h * kCOUT);
  float s = 0.f;
  for (int c = lane; c < kCOUT / 4; c += 32) {
    float4 wv = wrow[c];
    float4 av = arow[c];
    s += wv.x * av.x + wv.y * av.y + wv.z * av.z + wv.w * av.w;
  }
  s = wave_sum(s);
  if (lane == 0) W_att[k * kH + h] = s;
}

// ---- 2. a_d[n][h] = sum_k x[n][k] * W_att[k][h]   (one wave per vehicle row)
__global__ void k_ad(const float* __restrict__ x,
                     const float* __restrict__ W_att,
                     float* __restrict__ a_d) {
  int n = (blockIdx.x * blockDim.x + threadIdx.x) >> 5;
  int lane = threadIdx.x & 31;
  if (n >= kNV) return;
  const float4* xr = reinterpret_cast<const float4*>(x + (size_t)n * kCIN);
  const float4* wa = reinterpret_cast<const float4*>(W_att);
  float a0 = 0.f, a1 = 0.f, a2 = 0.f, a3 = 0.f;
  for (int k4 = lane; k4 < kCIN / 4; k4 += 32) {   // k4 indexes groups of 4 k's
    float4 xv = xr[k4];
    float4 w0 = wa[k4 * 4 + 0];
    float4 w1 = wa[k4 * 4 + 1];
    float4 w2 = wa[k4 * 4 + 2];
    float4 w3 = wa[k4 * 4 + 3];
    a0 += xv.x * w0.x + xv.y * w1.x + xv.z * w2.x + xv.w * w3.x;
    a1 += xv.x * w0.y + xv.y * w1.y + xv.z * w2.y + xv.w * w3.y;
    a2 += xv.x * w0.z + xv.y * w1.z + xv.z * w2.z + xv.w * w3.z;
    a3 += xv.x * w0.w + xv.y * w1.w + xv.z * w2.w + xv.w * w3.w;
  }
  a0 = wave_sum(a0); a1 = wave_sum(a1); a2 = wave_sum(a2); a3 = wave_sum(a3);
  if (lane == 0) {
    float* o = a_d + (size_t)n * 4;
    o[0] = a0; o[1] = a1; o[2] = a2; o[3] = a3;
  }
}

// ---- 3. h_src = CT_pad[32x2048] @ W via V_WMMA_F32_16X16X4_F32.
// CT_pad rows 20..31 are zero, so A loads are unconditional b64s (no exec-mask
// branching in the hot loop). Both 16-row M-tiles share the B fragment, so the
// 67MB W matrix streams exactly once. Output re-indexed as Hs[(h*20+cam)][c].
__global__ void k_hsrc(const float* __restrict__ CTp,
                       const float* __restrict__ W,
                       float* __restrict__ Hs) {
  int tileN = (blockIdx.x * blockDim.x + threadIdx.x) >> 5;  // 0..511
  int lane  = threadIdx.x & 31;
  int half  = lane >> 4, m = lane & 15;
  int col   = tileN * 16 + m;          // 0..8191 column of W
  int koff  = half * 2;
  const float* arow0 = CTp + (size_t)m * kCIN + koff;         // cam = m
  const float* arow1 = CTp + (size_t)(16 + m) * kCIN + koff;  // cam = 16+m (zero-padded)
  v8f acc0 = {}; v8f acc1 = {};
#pragma unroll 4
  for (int k = 0; k < kCIN; k += 4) {
    v2f b;
    b.x = W[(size_t)(k + koff)     * kNHC + col];
    b.y = W[(size_t)(k + koff + 1) * kNHC + col];
    v2f a0 = *reinterpret_cast<const v2f*>(arow0 + k);
    v2f a1 = *reinterpret_cast<const v2f*>(arow1 + k);
    acc0 = __builtin_amdgcn_wmma_f32_16x16x4_f32(false, a0, false, b, (short)0,
                                                 acc0, false, false);
    acc1 = __builtin_amdgcn_wmma_f32_16x16x4_f32(false, a1, false, b, (short)0,
                                                 acc1, false, false);
  }
  int h = col >> 11, c = col & (kCOUT - 1);
#pragma unroll
  for (int v = 0; v < 8; ++v) {
    int cam = v + half * 8;                       // D row: VGPR v -> M = v (+8 hi half)
    Hs[(size_t)(h * kNCAM + cam) * kCOUT + c] = acc0[v];
    int cam2 = cam + 16;
    if (cam2 < kNCAM)
      Hs[(size_t)(h * kNCAM + cam2) * kCOUT + c] = acc1[v];
  }
}

// ---- 4. a_s[cam][h] = sum_c Hs[h*20+cam][c] * att_src[h][c]
__global__ void k_as(const float* __restrict__ Hs,
                     const float* __restrict__ att_src,
                     float* __restrict__ a_s) {
  int wave = (blockIdx.x * blockDim.x + threadIdx.x) >> 5;
  int lane = threadIdx.x & 31;
  if (wave >= kNJ) return;
  int h = wave / kNCAM, cam = wave % kNCAM;
  const float4* hrow = reinterpret_cast<const float4*>(Hs + (size_t)wave * kCOUT);
  const float4* arow = reinterpret_cast<const float4*>(att_src + (size_t)h * kCOUT);
  float s = 0.f;
  for (int c = lane; c < kCOUT / 4; c += 32) {
    float4 hv = hrow[c];
    float4 av = arow[c];
    s += hv.x * av.x + hv.y * av.y + hv.z * av.z + hv.w * av.w;
  }
  s = wave_sum(s);
  if (lane == 0) a_s[cam * kH + h] = s;
}

// ---- 5. edge softmax (3 passes over E*H = 64K items)
__device__ __forceinline__ float edge_logit(const float* a_s, const float* a_d,
                                            int src, int dst, int h) {
  float v = a_s[src * kH + h] + a_d[dst * kH + h];
  return v > 0.f ? v : kNegSlope * v;
}

__global__ void k_edge_max(const int* __restrict__ esrc, const int* __restrict__ edst,
                           const float* __restrict__ a_s, const float* __restrict__ a_d,
                           unsigned* __restrict__ mEnc) {
  int t = blockIdx.x * blockDim.x + threadIdx.x;
  if (t >= kE * kH) return;
  int ei = t >> 2, h = t & 3;
  int src = esrc[ei], dst = edst[ei];
  float v = edge_logit(a_s, a_d, src, dst, h);
  atomicMax(&mEnc[dst * kH + h], f2ord(v));
}

__global__ void k_edge_den(const int* __restrict__ esrc, const int* __restrict__ edst,
                           const float* __restrict__ a_s, const float* __restrict__ a_d,
                           const unsigned* __restrict__ mEnc, float* __restrict__ den) {
  int t = blockIdx.x * blockDim.x + threadIdx.x;
  if (t >= kE * kH) return;
  int ei = t >> 2, h = t & 3;
  int src = esrc[ei], dst = edst[ei];
  float v = edge_logit(a_s, a_d, src, dst, h);
  float m = ord2f(mEnc[dst * kH + h]);
  atomicAdd(&den[dst * kH + h], __expf(v - m));
}

__global__ void k_edge_acc(const int* __restrict__ esrc, const int* __restrict__ edst,
                           const float* __restrict__ a_s, const float* __restrict__ a_d,
                           const unsigned* __restrict__ mEnc,
                           const float* __restrict__ den, float* __restrict__ S) {
  int t = blockIdx.x * blockDim.x + threadIdx.x;
  if (t >= kE * kH) return;
  int ei = t >> 2, h = t & 3;
  int src = esrc[ei], dst = edst[ei];
  float v = edge_logit(a_s, a_d, src, dst, h);
  float m = ord2f(mEnc[dst * kH + h]);
  float attn = __expf(v - m) / den[dst * kH + h];
  atomicAdd(&S[(size_t)dst * kNJ + h * kNCAM + src], attn);
}

// ---- 6. out = S[8192x80] @ Hs[80x2048] via WMMA f32, fused epilogue
//         d = x + alpha*(out/H + bias).  One wave per 16x16 tile, K=80 -> 20 WMMAs.
__global__ void k_out(const float* __restrict__ S, const float* __restrict__ Hs,
                      const float* __restrict__ x, const float* __restrict__ bias,
                      const float* __restrict__ alphap, float* __restrict__ out) {
  int w     = threadIdx.x >> 5;
  int lane  = threadIdx.x & 31;
  int half  = lane >> 4, m = lane & 15;
  int tileN = blockIdx.x;                 // 0..127
  int tileM = blockIdx.y * 8 + w;         // 0..511
  int col   = tileN * 16 + m;
  int rowA  = tileM * 16 + m;
  int koff  = half * 2;
  const float* srow = S + (size_t)rowA * kNJ + koff;
  v8f acc = {};
#pragma unroll
  for (int kk = 0; kk < kNJ; kk += 4) {
    v2f a = *reinterpret_cast<const v2f*>(srow + kk);
    v2f b;
    b.x = Hs[(size_t)(kk + koff)     * kCOUT + col];
    b.y = Hs[(size_t)(kk + koff + 1) * kCOUT + col];
    acc = __builtin_amdgcn_wmma_f32_16x16x4_f32(false, a, false, b, (short)0,
                                                acc, false, false);
  }
  float alpha = *alphap;
  float bcol  = bias[col];
#pragma unroll
  for (int v = 0; v < 8; ++v) {
    int r = tileM * 16 + v + half * 8;
    size_t idx = (size_t)r * kCOUT + col;
    out[idx] = x[idx] + alpha * (acc[v] * 0.25f + bcol);
  }
}

}  // namespace

extern "C" void kernel_launch(void* const* d_in, const int* in_sizes, int n_in,
                              void* d_out, int out_size, void* d_ws, size_t ws_size,
                              hipStream_t stream) {
  (void)in_sizes; (void)n_in; (void)out_size; (void)ws_size;
  const float* x       = (const float*)d_in[0];
  const float* ct      = (const float*)d_in[1];
  const float* W       = (const float*)d_in[2];
  const float* att_src = (const float*)d_in[3];
  const float* att_dst = (const float*)d_in[4];
  const float* bias    = (const float*)d_in[5];
  const float* alphap  = (const float*)d_in[6];
  // d_in[7] = unique_cams == arange(NCAM): gather is identity, ignored.
  const int* esrc = (const int*)d_in[8];
  const int* edst = (const int*)d_in[9];

  float* ws = (float*)d_ws;
  float*    Hs   = ws + OFF_HS;
  float*    Watt = ws + OFF_WATT;
  float*    a_d  = ws + OFF_AD;
  float*    a_s  = ws + OFF_AS;
  unsigned* mEnc = (unsigned*)(ws + OFF_MENC);
  float*    den  = ws + OFF_DEN;
  float*    S    = ws + OFF_S;
  float*    CTp  = ws + OFF_CTPAD;

  hipMemsetAsync(mEnc, 0, (size_t)kNV * kH * sizeof(unsigned), stream);  // ord-encoded -inf
  hipMemsetAsync(den,  0, (size_t)kNV * kH * sizeof(float), stream);
  hipMemsetAsync(S,    0, (size_t)kNV * kNJ * sizeof(float), stream);
  // zero-padded camera table: rows 20..31 = 0, rows 0..19 = camera_table
  hipMemsetAsync(CTp + (size_t)kNCAM * kCIN, 0,
                 (size_t)(32 - kNCAM) * kCIN * sizeof(float), stream);
  hipMemcpyAsync(CTp, ct, (size_t)kNCAM * kCIN * sizeof(float),
                 hipMemcpyDeviceToDevice, stream);

  k_fold_att<<<dim3((kCIN * kH) / 8), dim3(256), 0, stream>>>(W, att_dst, Watt);
  k_ad<<<dim3(kNV / 8), dim3(256), 0, stream>>>(x, Watt, a_d);
  k_hsrc<<<dim3((kNHC / 16) / 8), dim3(256), 0, stream>>>(CTp, W, Hs);
  k_as<<<dim3((kNJ + 7) / 8), dim3(256), 0, stream>>>(Hs, att_src, a_s);

  int et = kE * kH;
  k_edge_max<<<dim3((et + 255) / 256), dim3(256), 0, stream>>>(esrc, edst, a_s, a_d, mEnc);
  k_edge_den<<<dim3((et + 255) / 256), dim3(256), 0, stream>>>(esrc, edst, a_s, a_d, mEnc, den);
  k_edge_acc<<<dim3((et + 255) / 256), dim3(256), 0, stream>>>(esrc, edst, a_s, a_d, mEnc, den, S);

  k_out<<<dim3(kCOUT / 16, (kNV / 16) / 8), dim3(256), 0, stream>>>(
      S, Hs, x, bias, alphap, (float*)d_out);
}